// TransFusionHead_77060303225116
// MI455X (gfx1250) — compile-verified
//
#include <hip/hip_runtime.h>
#include <hip/hip_bf16.h>

// Problem constants (match reference)
#define BB 8
#define NN 4096
#define MM 1024
#define CC 10

typedef __attribute__((ext_vector_type(2))) float v2f;
typedef __attribute__((ext_vector_type(8))) float v8f;

// ---------------------------------------------------------------------------
// Kernel 1a: per-prediction tables.
//   diffP[B,N,16] : focal cost diff per class, padded to 16 (c>=10 -> 0)
//   boxA [B,N,8]  : 0.25*ax, 0.25*ay, xa1, ya1, xa2, ya2, area_a, 0
//   (REG_W = 0.25 folded into the normalized centers)
// ---------------------------------------------------------------------------
__global__ __launch_bounds__(256) void tf_precomp_pred(
    const float* __restrict__ bboxes,    // [B,N,7]
    const float* __restrict__ cls_pred,  // [B,N,C]
    float* __restrict__ diffP,           // [B,N,16]
    float* __restrict__ boxA)            // [B,N,8]
{
    int idx = blockIdx.x * blockDim.x + threadIdx.x;
    if (idx >= BB * NN) return;

    const float* cp = cls_pred + (size_t)idx * CC;
    float* dp = diffP + ((size_t)idx << 4);
#pragma unroll
    for (int c = 0; c < CC; ++c) {
        float x = cp[c];
        float p = 1.0f / (1.0f + __expf(-x));
        float omp = 1.0f - p;
        // neg = -log(1-p+eps)*(1-alpha)*p^2 ; pos = -log(p+eps)*alpha*(1-p)^2
        float neg = -__logf(omp + 1e-12f) * 0.75f * p * p;
        float pos = -__logf(p + 1e-12f) * 0.25f * omp * omp;
        dp[c] = (pos - neg) * 0.15f;     // CLS_W
    }
#pragma unroll
    for (int c = CC; c < 16; ++c) dp[c] = 0.0f;

    const float* bx = bboxes + (size_t)idx * 7;
    float x = bx[0], y = bx[1], dx = bx[3], dy = bx[4];
    float* oa = boxA + ((size_t)idx << 3);
    float xa1 = x - dx * 0.5f, ya1 = y - dy * 0.5f;
    float xa2 = x + dx * 0.5f, ya2 = y + dy * 0.5f;
    oa[0] = (x + 54.0f) * (0.25f / 108.0f);   // REG_W folded
    oa[1] = (y + 54.0f) * (0.25f / 108.0f);
    oa[2] = xa1; oa[3] = ya1; oa[4] = xa2; oa[5] = ya2;
    oa[6] = fmaxf(xa2 - xa1, 0.0f) * fmaxf(ya2 - ya1, 0.0f);
    oa[7] = 0.0f;
}

// ---------------------------------------------------------------------------
// Kernel 1b: per-GT tables.
//   boxB[B,M,8] : 0.25*bx, 0.25*by, xb1, yb1, xb2, yb2, area_b, 0
// ---------------------------------------------------------------------------
__global__ __launch_bounds__(256) void tf_precomp_gt(
    const float* __restrict__ gt_bboxes, // [B,M,7]
    float* __restrict__ boxB)            // [B,M,8]
{
    int idx = blockIdx.x * blockDim.x + threadIdx.x;
    if (idx >= BB * MM) return;

    const float* bx = gt_bboxes + (size_t)idx * 7;
    float x = bx[0], y = bx[1], dx = bx[3], dy = bx[4];
    float* ob = boxB + ((size_t)idx << 3);
    float xb1 = x - dx * 0.5f, yb1 = y - dy * 0.5f;
    float xb2 = x + dx * 0.5f, yb2 = y + dy * 0.5f;
    ob[0] = (x + 54.0f) * (0.25f / 108.0f);   // REG_W folded
    ob[1] = (y + 54.0f) * (0.25f / 108.0f);
    ob[2] = xb1; ob[3] = yb1; ob[4] = xb2; ob[5] = yb2;
    ob[6] = fmaxf(xb2 - xb1, 0.0f) * fmaxf(yb2 - yb1, 0.0f);
    ob[7] = 0.0f;
}

// ---------------------------------------------------------------------------
// Kernel 2: one wave32 per 16x32 output tile (two 16x16 m-subtiles sharing
// the A fragments and the per-row boxA loads).
//   cls via 2 x (3 x V_WMMA_F32_16X16X4_F32)  (diff[16x12] x onehot[12x16]),
//   then fused L1 + BEV-IoU in the C/D register layout.
//   All in-loop addresses are base + compile-time immediate offsets; bases
//   are 32-bit element offsets off uniform pointers (GVS addressing).
// ---------------------------------------------------------------------------
__global__ __launch_bounds__(256) void tf_cost_kernel(
    const float* __restrict__ diffP,   // [B,N,16]
    const float* __restrict__ boxA,    // [B,N,8]
    const float* __restrict__ boxB,    // [B,M,8]
    const int*   __restrict__ labels,  // [B,M]
    float* __restrict__ out)           // [B,N,M]
{
    const int lane  = threadIdx.x & 31;
    const int gwave = blockIdx.x * 8 + (threadIdx.x >> 5);

    const int TM = MM / 32;                 // 32 double-tiles along M
    const int tilesPerB = (NN / 16) * TM;   // 8192
    const int bat = gwave / tilesPerB;
    int rem = gwave - bat * tilesPerB;
    const int tn = rem / TM;
    const int tm = rem - tn * TM;
    const int n0 = tn * 16, m0 = tm * 32;

    const int col   = lane & 15;       // N(tile)-col for B/C/D, M-row for A
    const int khalf = lane >> 4;       // which K half / C-row half this lane holds

    // ---- A fragments (shared by both m-subtiles): one address, imm offsets ----
    const unsigned aOff = ((unsigned)(bat * NN + n0 + col) << 4) + 2u * khalf;
    v2f a0 = *(const v2f*)(diffP + aOff);
    v2f a1 = *(const v2f*)(diffP + aOff + 4);
    v2f a2 = *(const v2f*)(diffP + aOff + 8);

    // ---- B fragments: one-hot of gt labels, two m-subtiles ----
    const unsigned labOff = (unsigned)(bat * MM + m0 + col);
    const int labL = labels[labOff];
    const int labR = labels[labOff + 16];
    const int k0 = 2 * khalf;
    v2f bL0 = { (labL == k0    ) ? 1.0f : 0.0f, (labL == k0 + 1) ? 1.0f : 0.0f };
    v2f bL1 = { (labL == k0 + 4) ? 1.0f : 0.0f, (labL == k0 + 5) ? 1.0f : 0.0f };
    v2f bL2 = { (labL == k0 + 8) ? 1.0f : 0.0f, (labL == k0 + 9) ? 1.0f : 0.0f };
    v2f bR0 = { (labR == k0    ) ? 1.0f : 0.0f, (labR == k0 + 1) ? 1.0f : 0.0f };
    v2f bR1 = { (labR == k0 + 4) ? 1.0f : 0.0f, (labR == k0 + 5) ? 1.0f : 0.0f };
    v2f bR2 = { (labR == k0 + 8) ? 1.0f : 0.0f, (labR == k0 + 9) ? 1.0f : 0.0f };

    // Two independent accumulator chains (fills WMMA->WMMA hazard slots).
    v8f accL = {};
    v8f accR = {};
    accL = __builtin_amdgcn_wmma_f32_16x16x4_f32(false, a0, false, bL0, (short)0, accL, false, false);
    accR = __builtin_amdgcn_wmma_f32_16x16x4_f32(false, a0, false, bR0, (short)0, accR, false, false);
    accL = __builtin_amdgcn_wmma_f32_16x16x4_f32(false, a1, false, bL1, (short)0, accL, false, false);
    accR = __builtin_amdgcn_wmma_f32_16x16x4_f32(false, a1, false, bR1, (short)0, accR, false, false);
    accL = __builtin_amdgcn_wmma_f32_16x16x4_f32(false, a2, false, bL2, (short)0, accL, false, false);
    accR = __builtin_amdgcn_wmma_f32_16x16x4_f32(false, a2, false, bR2, (short)0, accR, false, false);

    // ---- per-m GT data: one address, imm offsets 0/16/512/528 bytes ----
    const unsigned bOff = (unsigned)(bat * MM + m0 + col) << 3;
    const float4 gL0 = *(const float4*)(boxB + bOff);
    const float4 gL1 = *(const float4*)(boxB + bOff + 4);
    const float4 gR0 = *(const float4*)(boxB + bOff + 128);
    const float4 gR1 = *(const float4*)(boxB + bOff + 132);

    // ---- loop bases: boxA rows advance by +8 floats, out rows by +MM ----
    const unsigned paOff  = (unsigned)(bat * NN + n0 + 8 * khalf) << 3;
    const unsigned outOff = (unsigned)(bat * NN + n0 + 8 * khalf) * MM + m0 + col;

#pragma unroll
    for (int r = 0; r < 8; ++r) {
        const float4 pa0 = *(const float4*)(boxA + paOff + r * 8);      // .25ax,.25ay,xa1,ya1
        const float4 pa1 = *(const float4*)(boxA + paOff + r * 8 + 4);  // xa2,ya2,area_a,0

        // left m-subtile
        {
            float reg = fabsf(pa0.x - gL0.x) + fabsf(pa0.y - gL0.y);
            float ix1 = fmaxf(pa0.z, gL0.z), iy1 = fmaxf(pa0.w, gL0.w);
            float ix2 = fminf(pa1.x, gL1.x), iy2 = fminf(pa1.y, gL1.y);
            float inter = fmaxf(ix2 - ix1, 0.0f) * fmaxf(iy2 - iy1, 0.0f);
            float uni = pa1.z + gL1.z - inter;
            float iou = inter * __builtin_amdgcn_rcpf(fmaxf(uni, 1e-6f));
            __builtin_nontemporal_store(fmaf(-0.25f, iou, accL[r] + reg),
                                        out + outOff + r * MM);
        }
        // right m-subtile
        {
            float reg = fabsf(pa0.x - gR0.x) + fabsf(pa0.y - gR0.y);
            float ix1 = fmaxf(pa0.z, gR0.z), iy1 = fmaxf(pa0.w, gR0.w);
            float ix2 = fminf(pa1.x, gR1.x), iy2 = fminf(pa1.y, gR1.y);
            float inter = fmaxf(ix2 - ix1, 0.0f) * fmaxf(iy2 - iy1, 0.0f);
            float uni = pa1.z + gR1.z - inter;
            float iou = inter * __builtin_amdgcn_rcpf(fmaxf(uni, 1e-6f));
            __builtin_nontemporal_store(fmaf(-0.25f, iou, accR[r] + reg),
                                        out + outOff + r * MM + 16);
        }
    }
}

// ---------------------------------------------------------------------------
extern "C" void kernel_launch(void* const* d_in, const int* in_sizes, int n_in,
                              void* d_out, int out_size, void* d_ws, size_t ws_size,
                              hipStream_t stream) {
    const float* bboxes    = (const float*)d_in[0];   // [B,N,7]
    const float* gt_bboxes = (const float*)d_in[1];   // [B,M,7]
    const float* cls_pred  = (const float*)d_in[2];   // [B,N,C]
    const int*   gt_labels = (const int*)d_in[3];     // [B,M]
    float* out = (float*)d_out;                       // [B,N,M]

    // Workspace layout (3.25 MB total)
    float* diffP = (float*)d_ws;                      // B*N*16
    float* boxA  = diffP + (size_t)BB * NN * 16;      // B*N*8
    float* boxB  = boxA  + (size_t)BB * NN * 8;       // B*M*8

    tf_precomp_pred<<<(BB * NN + 255) / 256, 256, 0, stream>>>(bboxes, cls_pred, diffP, boxA);
    tf_precomp_gt  <<<(BB * MM + 255) / 256, 256, 0, stream>>>(gt_bboxes, boxB);

    // One wave per 16x32 tile; 8 waves per block; all tiles full (EXEC all-1s).
    const int totalWaves = BB * (NN / 16) * (MM / 32); // 65536
    tf_cost_kernel<<<totalWaves / 8, 256, 0, stream>>>(diffP, boxA, boxB, gt_labels, out);
}